// Attention_42949672960723
// MI455X (gfx1250) — compile-verified
//
#include <hip/hip_runtime.h>

// Attention (Tacotron-style location-sensitive) for MI455X / gfx1250.
// B=32, T=2048, H=1024, K=11. Memory-bound: 2 passes over 256MB eh
// => ~515MB / 23.3 TB/s ~= 22us floor. All math kept in fp32.

#define B_  32
#define T_  2048
#define H_  1024
#define K_  11
#define PAD_ 5
#define TSPLIT_ 4
#define TCHUNK_ (T_ / TSPLIT_)   // 512

typedef __attribute__((ext_vector_type(2))) float v2f;
typedef __attribute__((ext_vector_type(8))) float v8f;

// ---------------------------------------------------------------------------
// Kernel 1: pax[b,t] = dot(eh[b,t,:], dhx[b,:]) via V_WMMA_F32_16X16X4_F32.
// One wave handles 16 consecutive t-rows (A = 16x4 f32 tile, K-loop over H).
// B-matrix = dhx chunk broadcast across all 16 N columns (staged in LDS), so
// every column of the 16x16 D accumulator equals the pax vector for the tile.
// Block = 256 threads = 8 waves -> 128 t-rows per block. Grid = (T/128, B).
// ---------------------------------------------------------------------------
__global__ __launch_bounds__(256) void pax_wmma_kernel(
    float* __restrict__ pax,          // [B,T] (scratch region of d_out)
    const float* __restrict__ eh,     // [B,T,H]
    const float* __restrict__ dhx)    // [B,H]
{
    const int b    = blockIdx.y;
    const int tblk = blockIdx.x;
    const int tid  = threadIdx.x;

    __shared__ float s_d[H_];
    #pragma unroll
    for (int i = tid; i < H_; i += 256) s_d[i] = dhx[b * H_ + i];
    __syncthreads();

    const int wave = tid >> 5;
    const int lane = tid & 31;
    const int l16  = lane & 15;        // M row within tile
    const int hi   = lane >> 4;        // K half: lanes 0-15 -> K0/K1, 16-31 -> K2/K3
    const int t0   = tblk * 128 + wave * 16;

    // A fragment source: eh row for (b, t0+l16), offset by 2*hi floats so the
    // float2 at +k0 delivers (K = k0+2*hi, k0+2*hi+1) per the ISA A layout.
    const float* __restrict__ row =
        eh + ((size_t)b * T_ + (size_t)(t0 + l16)) * H_ + 2 * hi;
    const float* __restrict__ dch = &s_d[2 * hi];

    v8f c = {};
    #pragma unroll 4
    for (int k0 = 0; k0 < H_; k0 += 4) {
        v2f a  = *(const v2f*)(row + k0);   // global_load_b64, 16B/row contiguous
        v2f bv = *(const v2f*)(dch + k0);   // ds_load_2addr_b64, lane broadcast
        // 8 args: (neg_a, A, neg_b, B, c_mod, C, reuse_a, reuse_b)
        c = __builtin_amdgcn_wmma_f32_16x16x4_f32(
                false, a, false, bv, (short)0, c, false, false);
    }

    // C/D layout: VGPR r -> M=r (lanes 0-15) / M=r+8 (lanes 16-31); all N
    // columns are identical, so lane 0 and lane 16 hold the full pax tile.
    if (l16 == 0) {
        float* __restrict__ out = pax + (size_t)b * T_ + t0 + hi * 8;
        #pragma unroll
        for (int r = 0; r < 8; ++r) out[r] = c[r];
    }
}

// ---------------------------------------------------------------------------
// Kernel 2: fused location-conv + bias + softmax over T, in place on pax.
// One block (256 threads, 8 waves) per batch. ax row + halo staged in LDS.
// ---------------------------------------------------------------------------
__global__ __launch_bounds__(256) void conv_softmax_kernel(
    float* __restrict__ paxn,         // in: pax[b,T], out: ax_new[b,T]
    const float* __restrict__ ax,     // [B,T]
    const float* __restrict__ w,      // [K]
    const float* __restrict__ cb)     // [1]
{
    const int b   = blockIdx.x;
    const int tid = threadIdx.x;

    __shared__ float s_ax[T_ + 2 * PAD_];
    __shared__ float s_w[K_];
    __shared__ float s_red[8];

    for (int i = tid; i < T_ + 2 * PAD_; i += 256) {
        int t = i - PAD_;
        s_ax[i] = (t >= 0 && t < T_) ? ax[b * T_ + t] : 0.0f;
    }
    if (tid < K_) s_w[tid] = w[tid];
    __syncthreads();

    const float bias = cb[0];
    float v[8];
    float m = -3.402823466e+38f;
    #pragma unroll
    for (int i = 0; i < 8; ++i) {
        const int t = tid + i * 256;
        float loc = bias;
        #pragma unroll
        for (int k = 0; k < K_; ++k) loc = fmaf(s_ax[t + k], s_w[k], loc);
        float s = paxn[b * T_ + t] + loc;
        v[i] = s;
        m = fmaxf(m, s);
    }

    // max reduce: wave32 shuffle, then cross-wave via LDS
    #pragma unroll
    for (int off = 16; off > 0; off >>= 1)
        m = fmaxf(m, __shfl_xor(m, off, 32));
    if ((tid & 31) == 0) s_red[tid >> 5] = m;
    __syncthreads();
    float gm = s_red[0];
    #pragma unroll
    for (int j = 1; j < 8; ++j) gm = fmaxf(gm, s_red[j]);
    __syncthreads();

    float sum = 0.0f;
    #pragma unroll
    for (int i = 0; i < 8; ++i) {
        v[i] = expf(v[i] - gm);
        sum += v[i];
    }
    #pragma unroll
    for (int off = 16; off > 0; off >>= 1)
        sum += __shfl_xor(sum, off, 32);
    if ((tid & 31) == 0) s_red[tid >> 5] = sum;
    __syncthreads();
    float gs = 0.0f;
    #pragma unroll
    for (int j = 0; j < 8; ++j) gs += s_red[j];
    const float inv = 1.0f / gs;

    #pragma unroll
    for (int i = 0; i < 8; ++i)
        paxn[b * T_ + tid + i * 256] = v[i] * inv;
}

// ---------------------------------------------------------------------------
// Kernel 3a (preferred, split-T): partial[tc,b,h] = sum over one T-chunk.
// Grid = (H/256, B, TSPLIT) = 512 workgroups -> enough parallelism to keep
// every WGP's HBM path busy during the second 256MB pass.
// ---------------------------------------------------------------------------
__global__ __launch_bounds__(256) void sx_partial_kernel(
    float* __restrict__ part,         // [TSPLIT,B,H]
    const float* __restrict__ axn,    // [B,T]
    const float* __restrict__ eh)     // [B,T,H]
{
    const int b   = blockIdx.y;
    const int hc  = blockIdx.x;
    const int tc  = blockIdx.z;
    const int tid = threadIdx.x;
    const int h   = hc * 256 + tid;
    const int tbase = tc * TCHUNK_;

    __shared__ float s_a[TCHUNK_];
    for (int i = tid; i < TCHUNK_; i += 256) s_a[i] = axn[b * T_ + tbase + i];
    __syncthreads();

    const float* __restrict__ p =
        eh + ((size_t)b * T_ + tbase) * H_ + h;

    float a0 = 0.0f, a1 = 0.0f, a2 = 0.0f, a3 = 0.0f;
    for (int t = 0; t < TCHUNK_; t += 8) {
        if (t + 16 < TCHUNK_) __builtin_prefetch(p + (size_t)(t + 16) * H_, 0, 0);
        a0 = fmaf(s_a[t + 0], p[(size_t)(t + 0) * H_], a0);
        a1 = fmaf(s_a[t + 1], p[(size_t)(t + 1) * H_], a1);
        a2 = fmaf(s_a[t + 2], p[(size_t)(t + 2) * H_], a2);
        a3 = fmaf(s_a[t + 3], p[(size_t)(t + 3) * H_], a3);
        a0 = fmaf(s_a[t + 4], p[(size_t)(t + 4) * H_], a0);
        a1 = fmaf(s_a[t + 5], p[(size_t)(t + 5) * H_], a1);
        a2 = fmaf(s_a[t + 6], p[(size_t)(t + 6) * H_], a2);
        a3 = fmaf(s_a[t + 7], p[(size_t)(t + 7) * H_], a3);
    }
    part[((size_t)tc * B_ + b) * H_ + h] = (a0 + a1) + (a2 + a3);
}

__global__ __launch_bounds__(256) void sx_reduce_kernel(
    float* __restrict__ sx,           // [B,H]
    const float* __restrict__ part)   // [TSPLIT,B,H]
{
    const int i = blockIdx.x * 256 + threadIdx.x;   // i < B*H
    const size_t BH = (size_t)B_ * H_;
    float s = part[i];
    #pragma unroll
    for (int tc = 1; tc < TSPLIT_; ++tc) s += part[tc * BH + i];
    sx[i] = s;
}

// ---------------------------------------------------------------------------
// Kernel 3b (fallback if workspace too small): single-pass weighted sum.
// ---------------------------------------------------------------------------
__global__ __launch_bounds__(256) void sx_kernel(
    float* __restrict__ sx,           // [B,H]
    const float* __restrict__ axn,    // [B,T]
    const float* __restrict__ eh)     // [B,T,H]
{
    const int b   = blockIdx.y;
    const int hc  = blockIdx.x;
    const int tid = threadIdx.x;
    const int h   = hc * 256 + tid;

    __shared__ float s_a[T_];
    for (int i = tid; i < T_; i += 256) s_a[i] = axn[b * T_ + i];
    __syncthreads();

    const float* __restrict__ p = eh + (size_t)b * T_ * H_ + h;

    float a0 = 0.0f, a1 = 0.0f, a2 = 0.0f, a3 = 0.0f;
    for (int t = 0; t < T_; t += 8) {
        if (t + 16 < T_) __builtin_prefetch(p + (size_t)(t + 16) * H_, 0, 0);
        a0 = fmaf(s_a[t + 0], p[(size_t)(t + 0) * H_], a0);
        a1 = fmaf(s_a[t + 1], p[(size_t)(t + 1) * H_], a1);
        a2 = fmaf(s_a[t + 2], p[(size_t)(t + 2) * H_], a2);
        a3 = fmaf(s_a[t + 3], p[(size_t)(t + 3) * H_], a3);
        a0 = fmaf(s_a[t + 4], p[(size_t)(t + 4) * H_], a0);
        a1 = fmaf(s_a[t + 5], p[(size_t)(t + 5) * H_], a1);
        a2 = fmaf(s_a[t + 6], p[(size_t)(t + 6) * H_], a2);
        a3 = fmaf(s_a[t + 7], p[(size_t)(t + 7) * H_], a3);
    }
    sx[b * H_ + h] = (a0 + a1) + (a2 + a3);
}

// ---------------------------------------------------------------------------
extern "C" void kernel_launch(void* const* d_in, const int* in_sizes, int n_in,
                              void* d_out, int out_size, void* d_ws, size_t ws_size,
                              hipStream_t stream)
{
    (void)in_sizes; (void)n_in; (void)out_size;

    const float* eh  = (const float*)d_in[0];   // [B,T,H]
    const float* dhx = (const float*)d_in[1];   // [B,1,H]
    const float* ax  = (const float*)d_in[2];   // [B,T]
    const float* w   = (const float*)d_in[3];   // [1,1,K]
    const float* cb  = (const float*)d_in[4];   // [1]

    float* sx  = (float*)d_out;                   // [B,1,H] = 32768 floats
    float* axn = (float*)d_out + (size_t)B_ * H_; // [B,T]   = 65536 floats
                                                  // (pax scratch, then ax_new)

    pax_wmma_kernel    <<<dim3(T_ / 128, B_), 256, 0, stream>>>(axn, eh, dhx);
    conv_softmax_kernel<<<B_,                 256, 0, stream>>>(axn, ax, w, cb);

    const size_t part_bytes = (size_t)TSPLIT_ * B_ * H_ * sizeof(float);
    if (ws_size >= part_bytes) {
        float* part = (float*)d_ws;
        sx_partial_kernel<<<dim3(H_ / 256, B_, TSPLIT_), 256, 0, stream>>>(part, axn, eh);
        sx_reduce_kernel <<<(B_ * H_) / 256,             256, 0, stream>>>(sx, part);
    } else {
        sx_kernel        <<<dim3(H_ / 256, B_),          256, 0, stream>>>(sx, axn, eh);
    }
}